// PrePrompt_55937654063433
// MI455X (gfx1250) — compile-verified
//
#include <hip/hip_runtime.h>

typedef __attribute__((ext_vector_type(2))) float v2f;
typedef __attribute__((ext_vector_type(8))) float v8f;

#define NG 6
#define NN 20000
#define NF 512
#define NE 640000
#define NT (NN / 16)   // 1250 node tiles
#define FT (NF / 16)   // 32 feature tiles

// ---------------------------------------------------------------------------
// Kernel 1: per-wave 16x16 f32 tile.
//   out[g,n,0:F]   = relu(seq + w_pre[g]) + w_sum + w_bal[g,0:F]
//   out[g,n,F:2F]  = w_bal[g,F:2F]            (init for edge scatter)
// Broadcast row-vector adds are done as rank-1 WMMA updates:
//   D = A(16x4, first col = 1) x B(4x16, first row = w) + C  ==  C + w[n]
// C/D layout (ISA 7.12.2): lanes 0-15 -> M=r (VGPR r), N=lane;
//                          lanes 16-31 -> M=8+r, N=lane-16.
// We map M -> node, N -> feature.
// ---------------------------------------------------------------------------
__global__ __launch_bounds__(256) void pre_tile_kernel(
    const float* __restrict__ seqs,   // [G,N,F]
    const float* __restrict__ w_pre,  // [G,F]
    const float* __restrict__ w_sum,  // [F]
    const float* __restrict__ w_bal,  // [G,2F]
    float* __restrict__ out)          // [G,N,2F]
{
  const int wave = threadIdx.x >> 5;
  const int lane = threadIdx.x & 31;
  const int tile = blockIdx.x * 8 + wave;          // grid sized exactly
  const int g    = tile / (NT * FT);
  const int rem  = tile % (NT * FT);
  const int nt   = rem / FT;
  const int ft   = rem % FT;

  const int f  = ft * 16 + (lane & 15);            // feature column (N)
  const int nb = nt * 16 + ((lane >> 4) << 3);     // node base (M) for this half-wave

  // Load 16x16 seq tile into WMMA C layout: 8 column-strided b32 loads.
  const float* sp = seqs + ((size_t)g * NN + nb) * NF + f;
  v8f c;
  #pragma unroll
  for (int r = 0; r < 8; ++r) c[r] = sp[(size_t)r * NF];

  const float wp  = w_pre[g * NF + f];
  const float wsb = w_sum[f] + w_bal[(size_t)g * 2 * NF + f];
  const float wbh = w_bal[(size_t)g * 2 * NF + NF + f];

  // A: 16x4, A[m,0]=1 else 0.  Lanes 0-15 hold K=0,1; lanes 16-31 hold K=2,3.
  const bool hi = lane >= 16;
  v2f a;  a.x  = hi ? 0.0f : 1.0f;  a.y  = 0.0f;
  v2f b1; b1.x = hi ? 0.0f : wp;    b1.y = 0.0f;   // B[0,n] = w_pre[n]
  v2f b2; b2.x = hi ? 0.0f : wsb;   b2.y = 0.0f;   // B[0,n] = w_sum+w_bal_lo

  // seq + w_pre (rank-1 broadcast via matrix core)
  c = __builtin_amdgcn_wmma_f32_16x16x4_f32(false, a, false, b1,
                                            (short)0, c, false, false);
  #pragma unroll
  for (int r = 0; r < 8; ++r) c[r] = fmaxf(c[r], 0.0f);  // relu
  // + (w_sum + w_bal_lo)
  c = __builtin_amdgcn_wmma_f32_16x16x4_f32(false, a, false, b2,
                                            (short)0, c, false, false);

  float* op = out + ((size_t)g * NN + nb) * (2 * NF) + f;
  #pragma unroll
  for (int r = 0; r < 8; ++r) {
    // first half: written once, never re-read on-chip -> non-temporal
    __builtin_nontemporal_store(c[r], op + (size_t)r * (2 * NF));
    // second half: bias init; will be hammered by L2 atomics -> regular
    op[(size_t)r * (2 * NF) + NF] = wbh;
  }
}

// ---------------------------------------------------------------------------
// Kernel 2: edge scatter. 128 threads per edge, float4 of features each.
//   out[g, rows[e], F+f] += vals[e] * (relu(seq[g,cols[e],f]+w_pre)+w_sum)
// pre is recomputed from seq (saves a 246 MB intermediate); gathers hit L2
// (40 MB/graph working set); scatter uses global_atomic_add_f32 (no-return).
// ---------------------------------------------------------------------------
__global__ __launch_bounds__(256) void edge_scatter_kernel(
    const float* __restrict__ seqs,
    const int*   __restrict__ rows,
    const int*   __restrict__ cols,
    const float* __restrict__ vals,
    const float* __restrict__ w_pre,
    const float* __restrict__ w_sum,
    float* __restrict__ out)
{
  const long long tid = (long long)blockIdx.x * blockDim.x + threadIdx.x;
  const long long eg  = tid >> 7;          // global edge id (wave-uniform)
  const int       q   = (int)(tid & 127);  // float4 chunk: features q*4..q*4+3
  if (eg >= (long long)NG * NE) return;

  const int g = (int)(eg / NE);
  const int e = (int)(eg % NE);
  const size_t eb = (size_t)g * NE + e;
  const int   row = rows[eb];
  const int   col = cols[eb];
  const float v   = vals[eb];
  const int   f   = q * 4;

  const float4 s  = *(const float4*)(seqs + ((size_t)g * NN + col) * NF + f);
  const float4 wp = *(const float4*)(w_pre + (size_t)g * NF + f);
  const float4 ws = *(const float4*)(w_sum + f);

  float4 p;
  p.x = v * (fmaxf(s.x + wp.x, 0.0f) + ws.x);
  p.y = v * (fmaxf(s.y + wp.y, 0.0f) + ws.y);
  p.z = v * (fmaxf(s.z + wp.z, 0.0f) + ws.z);
  p.w = v * (fmaxf(s.w + wp.w, 0.0f) + ws.w);

  float* dst = out + ((size_t)g * NN + row) * (2 * NF) + NF + f;
  unsafeAtomicAdd(dst + 0, p.x);
  unsafeAtomicAdd(dst + 1, p.y);
  unsafeAtomicAdd(dst + 2, p.z);
  unsafeAtomicAdd(dst + 3, p.w);
}

extern "C" void kernel_launch(void* const* d_in, const int* in_sizes, int n_in,
                              void* d_out, int out_size, void* d_ws, size_t ws_size,
                              hipStream_t stream) {
  const float* seqs  = (const float*)d_in[0];
  const int*   rows  = (const int*)  d_in[1];
  const int*   cols  = (const int*)  d_in[2];
  const float* vals  = (const float*)d_in[3];
  const float* w_pre = (const float*)d_in[4];
  const float* w_sum = (const float*)d_in[5];
  const float* w_bal = (const float*)d_in[6];
  float* out = (float*)d_out;

  // Kernel 1: one wave per 16x16 tile; 8 waves per block.
  const int tiles  = NG * NT * FT;        // 240000
  const int blocks1 = tiles / 8;          // 30000
  pre_tile_kernel<<<blocks1, 256, 0, stream>>>(seqs, w_pre, w_sum, w_bal, out);

  // Kernel 2: 128 threads/edge, 2 edges per 256-thread block.
  const long long work = (long long)NG * NE * 128;   // 491.52M threads
  const int blocks2 = (int)(work / 256);             // 1,920,000
  edge_scatter_kernel<<<blocks2, 256, 0, stream>>>(seqs, rows, cols, vals,
                                                   w_pre, w_sum, out);
}